// CustomConvolutionLayer_12515534701154
// MI455X (gfx1250) — compile-verified
//
#include <hip/hip_runtime.h>
#include <stdint.h>

// ---------------------------------------------------------------------------
// 3x3 VALID conv of 64 x 1024x1024 f32 images, bandwidth-bound (~23us floor).
// Data movement: TDM tensor_load_to_lds double-buffered tiles (TENSORcnt),
// one workgroup barrier per tile. Math: V_WMMA_F32_16X16X4_F32 with
// precomputed Toeplitz B operands. NT stores keep output out of L2.
// ---------------------------------------------------------------------------

typedef float    v2f   __attribute__((ext_vector_type(2)));
typedef float    v8f   __attribute__((ext_vector_type(8)));
typedef uint32_t u32x4 __attribute__((ext_vector_type(4)));
typedef int      i32x4 __attribute__((ext_vector_type(4)));
typedef int      i32x8 __attribute__((ext_vector_type(8)));

#define IMG_D     1024
#define OUT_D     1022
#define OUT_PIX   (OUT_D * OUT_D)
#define TILE_ROWS 16      // output rows per row-tile
#define TILE_COLS 128     // output cols per workgroup (8 waves x 16)
#define IN_ROWS   18      // TILE_ROWS + 2 taps
#define IN_COLS   132     // TILE_COLS + 2 taps, padded to K=20 per-wave window
#define ROW_TILES 8       // row-tiles per workgroup (grid.y covers 8 groups)

// Issue one 2D TDM load: tile IN_COLS x IN_ROWS dwords from a 1024x1024 f32
// tensor into LDS at lds_off. OOB rows/cols (image edge) handled by tensor
// dims; any element that could be garbage only feeds masked outputs or
// Toeplitz-zero columns.
__device__ __forceinline__ void tdm_load_tile(uint32_t lds_off, const float* gptr) {
    uint64_t ga = (uint64_t)(uintptr_t)gptr;
    u32x4 g0;
    g0[0] = 1u;                                               // count=1, user D#
    g0[1] = lds_off;                                          // lds_addr
    g0[2] = (uint32_t)ga;                                     // global_addr[31:0]
    g0[3] = (uint32_t)((ga >> 32) & 0x01FFFFFFu) | (2u << 30);// addr[56:32] | type=2
    i32x8 g1;
    g1[0] = (int)(2u << 16);                                  // data_size=4B, no mask
    g1[1] = (int)((uint32_t)(IMG_D & 0xFFFF) << 16);          // tensor_dim0 lo16
    g1[2] = (int)(((uint32_t)(IMG_D >> 16)) |
                  ((uint32_t)(IMG_D & 0xFFFF) << 16));        // dim0 hi16 | dim1 lo16
    g1[3] = (int)(((uint32_t)(IMG_D >> 16)) |
                  ((uint32_t)IN_COLS << 16));                 // dim1 hi16 | tile_dim0
    g1[4] = (int)IN_ROWS;                                     // tile_dim1 (tile_dim2=0)
    g1[5] = (int)IMG_D;                                       // tensor_dim0_stride lo32
    g1[6] = (int)(((uint32_t)((IMG_D * IMG_D) & 0xFFFF)) << 16); // stride0 hi | stride1 lo
    g1[7] = (int)((uint32_t)(IMG_D * IMG_D) >> 16);           // stride1 hi
    i32x4 g2 = {0, 0, 0, 0};
    i32x4 g3 = {0, 0, 0, 0};
    i32x8 g4 = {0, 0, 0, 0, 0, 0, 0, 0};                      // unused extra group
    __builtin_amdgcn_tensor_load_to_lds(g0, g1, g2, g3, g4, 0);
}

__global__ __launch_bounds__(256) void conv3x3_tdm_wmma(const float* __restrict__ x,
                                                        const float* __restrict__ kw,
                                                        float* __restrict__ out) {
    __shared__ float smem[2][IN_ROWS * IN_COLS];

    const int colTile = blockIdx.x;       // 0..7  (128-col strip)
    const int rowGrp  = blockIdx.y;       // 0..7  (8 row-tiles each)
    const int img     = blockIdx.z;       // 0..63
    const int col0    = colTile * TILE_COLS;

    const int tid    = threadIdx.x;
    const int lane   = tid & 31;
    const int waveId = tid >> 5;          // 0..7
    const int n0w    = waveId * 16;       // wave's 16-col window in the tile
    const int nIdx   = lane & 15;         // N (B/C/D) or M (A) index
    const int half   = lane >> 4;         // lane half selects K pair / M+8

    // Scalar (SGPR) copies of wave-uniform facts -> real s_cbranch, EXEC
    // untouched (TDM issues regardless of EXEC; WMMA needs EXEC all-ones).
    const int waveIdS = __builtin_amdgcn_readfirstlane(waveId);
    const int n0wS    = __builtin_amdgcn_readfirstlane(n0w);

    // Weights (uniform scalar loads)
    float w0[3], w1[3], w2[3];
    #pragma unroll
    for (int i = 0; i < 3; ++i) { w0[i] = kw[i]; w1[i] = kw[3 + i]; w2[i] = kw[6 + i]; }

    // Precompute the 15 Toeplitz B operands (4x16 each), invariant across tiles.
    // B layout: VGPR0 holds K rows {0,2}, VGPR1 holds K rows {1,3}; N = lane%16.
    // T_dy[c, n] = w[dy][c - n] if 0 <= c-n <= 2 else 0, c = 4*kc + K.
    v2f B[3][5];
    #pragma unroll
    for (int dy = 0; dy < 3; ++dy) {
        const float* wr = (dy == 0) ? w0 : (dy == 1) ? w1 : w2;
        #pragma unroll
        for (int kc = 0; kc < 5; ++kc) {
            int c0 = 4 * kc + 2 * half;
            int d0 = c0 - nIdx, d1 = c0 + 1 - nIdx;
            v2f b;
            b.x = (d0 == 0) ? wr[0] : (d0 == 1) ? wr[1] : (d0 == 2) ? wr[2] : 0.f;
            b.y = (d1 == 0) ? wr[0] : (d1 == 1) ? wr[1] : (d1 == 2) ? wr[2] : 0.f;
            B[dy][kc] = b;
        }
    }

    const float* imgBase = x + (size_t)img * (IMG_D * IMG_D);

    // Prologue: stage first tile (wave 0 only, scalar branch).
    if (waveIdS == 0) {
        tdm_load_tile((uint32_t)(uintptr_t)&smem[0][0],
                      imgBase + (size_t)((rowGrp * ROW_TILES) * TILE_ROWS) * IMG_D + col0);
    }

    int buf = 0;
    #pragma unroll 1
    for (int rt = 0; rt < ROW_TILES; ++rt) {
        const int row0 = (rowGrp * ROW_TILES + rt) * TILE_ROWS;

        if (waveIdS == 0) __builtin_amdgcn_s_wait_tensorcnt(0); // tile[buf] landed
        __syncthreads();   // (a) tile[buf] visible, (b) all waves done with buf^1

        // Prefetch next tile into buf^1, overlapped with compute below.
        if (waveIdS == 0 && rt + 1 < ROW_TILES) {
            tdm_load_tile((uint32_t)(uintptr_t)&smem[buf ^ 1][0],
                          imgBase + (size_t)(row0 + TILE_ROWS) * IMG_D + col0);
        }

        // Implicit-Toeplitz GEMM: 3 (dy) x 5 (K chunks) WMMA f32 16x16x4.
        // A lane L: M = L%16, K = 2*(L>=16) + {0,1}  -> one ds_load_b64/chunk.
        const float* tile = &smem[buf][0];
        v8f acc = {};
        #pragma unroll
        for (int dy = 0; dy < 3; ++dy) {
            const float* rowp = tile + (nIdx + dy) * IN_COLS + n0w + 2 * half;
            #pragma unroll
            for (int kc = 0; kc < 5; ++kc) {
                v2f a = *(const v2f*)(rowp + 4 * kc);
                acc = __builtin_amdgcn_wmma_f32_16x16x4_f32(
                    false, a, false, B[dy][kc], (short)0, acc, false, false);
            }
        }

        // D layout: VGPR v -> M = v + 8*half, N = nIdx.
        // Interior tiles (all but last row-tile / last wave of last strip):
        // unmasked NT stores on a scalar branch; edges take the masked path.
        const int ocol  = col0 + n0w + nIdx;
        const int orow0 = row0 + 8 * half;
        float* op = out + (size_t)img * OUT_PIX + (size_t)orow0 * OUT_D + ocol;
        const int interior = __builtin_amdgcn_readfirstlane(
            ((row0 + TILE_ROWS <= OUT_D) && (col0 + n0wS + 15 < OUT_D)) ? 1 : 0);
        if (interior) {
            #pragma unroll
            for (int v = 0; v < 8; ++v)
                __builtin_nontemporal_store(acc[v], op + (size_t)v * OUT_D);
        } else if (ocol < OUT_D) {
            #pragma unroll
            for (int v = 0; v < 8; ++v)
                if (orow0 + v < OUT_D) op[(size_t)v * OUT_D] = acc[v];
        }

        buf ^= 1;
    }
}

extern "C" void kernel_launch(void* const* d_in, const int* in_sizes, int n_in,
                              void* d_out, int out_size, void* d_ws, size_t ws_size,
                              hipStream_t stream) {
    (void)in_sizes; (void)n_in; (void)d_ws; (void)ws_size; (void)out_size;
    const float* x  = (const float*)d_in[0];   // [64, 1024*1024] f32
    const float* kw = (const float*)d_in[1];   // [3, 3] f32
    float* out = (float*)d_out;                // [64, 1022*1022] f32

    dim3 grid(IMG_D / TILE_COLS,               // 8 column strips
              (IMG_D / TILE_ROWS) / ROW_TILES, // 8 row groups
              64);                             // images
    conv3x3_tdm_wmma<<<grid, dim3(256), 0, stream>>>(x, kw, out);
}